// TargetAssigner_90675349553799
// MI455X (gfx1250) — compile-verified
//
#include <hip/hip_runtime.h>
#include <hip/hip_bf16.h>
#include <math.h>

// Problem constants from the reference setup
#define BB 4
#define NN 16384
#define MM 256
#define AA 3
#define MARGIN 1e-2f   // |dist^2 - r^2| band where we fall back to exact test

typedef float v2f __attribute__((ext_vector_type(2)));
typedef float v8f __attribute__((ext_vector_type(8)));

// One block = 256 threads = 8 waves (wave32). Each wave handles 16 keypoints
// against all 256 boxes via V_WMMA_F32_16X16X4_F32 (rank-4 GEMM trick:
// D = ||c||^2 - 2 c.k ; dist^2 = D + ||k||^2).
__global__ __launch_bounds__(256) void TargetAssigner_kernel(
    const float* __restrict__ keypoints,     // [B,N,3]
    const float* __restrict__ boxes,         // [M,7]
    const int*   __restrict__ class_ids,     // [M]
    const int*   __restrict__ box_batch,     // [M]
    const float* __restrict__ anchor_sizes,  // [A,3]
    const float* __restrict__ anchor_radii,  // [A]
    float* __restrict__ out_cls,             // [B,N,A+1]
    float* __restrict__ out_reg)             // [B,N,A+1,7]
{
    __shared__ __align__(16) float4 sA[MM];       // (cx, cy, cz, ||c||^2) -> WMMA A
    __shared__ __align__(16) float  sR2[MM];      // radius^2 per box
    __shared__ __align__(16) float  sRad[MM];     // radius per box (exact recheck)
    __shared__ __align__(16) int    sMeta[MM];    // class | (batch << 2)
    __shared__ __align__(16) float  sBox[MM * 7]; // raw box rows for regression
    __shared__ float sAS[AA * 3];                 // anchor sizes

    const int tid = threadIdx.x;

    // ---- stage box-derived data into LDS (one box per thread) ----
    {
        const int m = tid;  // 256 threads == 256 boxes
        float cx = boxes[m * 7 + 0];
        float cy = boxes[m * 7 + 1];
        float cz = boxes[m * 7 + 2];
        #pragma unroll
        for (int i = 0; i < 7; ++i) sBox[m * 7 + i] = boxes[m * 7 + i];
        int cl = class_ids[m];
        int bb = box_batch[m];
        float r = anchor_radii[cl];
        sRad[m]  = r;
        sR2[m]   = r * r;
        sMeta[m] = cl | (bb << 2);
        float4 a; a.x = cx; a.y = cy; a.z = cz; a.w = cx * cx + cy * cy + cz * cz;
        sA[m] = a;
        if (tid < AA * 3) sAS[tid] = anchor_sizes[tid];
    }
    __syncthreads();

    const int  lane = tid & 31;
    const int  wave = tid >> 5;
    const bool hi   = lane >= 16;   // lane-half: selects K=2,3 (A/B) and rows +8 (D)
    const int  col  = lane & 15;    // keypoint within the wave's 16-keypoint tile

    const int g  = (blockIdx.x * 8 + wave) * 16 + col; // flat keypoint id in [0, B*N)
    const int b4 = (g >> 14) << 2;                     // batch index << 2 (N = 16384)

    const float kx = keypoints[g * 3 + 0];
    const float ky = keypoints[g * 3 + 1];
    const float kz = keypoints[g * 3 + 2];
    const float kk = kx * kx + ky * ky + kz * kz;

    // B operand: column = [-2kx, -2ky, -2kz, 1]^T, K split across lane halves
    v2f bmat;
    bmat.x = hi ? (-2.0f * kz) : (-2.0f * kx);
    bmat.y = hi ? 1.0f         : (-2.0f * ky);

    int best0 = -1, best1 = -1, best2 = -1;

    // partial unroll: 4 static WMMAs, keeps VGPR count < 128 for high occupancy
    #pragma unroll 4
    for (int t = 0; t < 16; ++t) {
        const int mbase = t * 16;
        // A operand: row m = mbase+col is [cx, cy, cz, ||c||^2], K split over halves
        const float4 av = sA[mbase + col];
        v2f amat;
        amat.x = hi ? av.z : av.x;
        amat.y = hi ? av.w : av.y;

        v8f acc = {0.f, 0.f, 0.f, 0.f, 0.f, 0.f, 0.f, 0.f};
        acc = __builtin_amdgcn_wmma_f32_16x16x4_f32(
            /*neg_a=*/false, amat, /*neg_b=*/false, bmat,
            /*c_mod=*/(short)0, acc, /*reuse_a=*/false, /*reuse_b=*/false);

        const int rbase = mbase + (hi ? 8 : 0);  // D rows owned by this lane

        // vectorized metadata for the 8 rows: 2x ds_load_b128 each
        const float4 r2lo = ((const float4*)&sR2[rbase])[0];
        const float4 r2hi = ((const float4*)&sR2[rbase])[1];
        const int4   mtlo = ((const int4*)&sMeta[rbase])[0];
        const int4   mthi = ((const int4*)&sMeta[rbase])[1];
        const float r2v[8] = {r2lo.x, r2lo.y, r2lo.z, r2lo.w,
                              r2hi.x, r2hi.y, r2hi.z, r2hi.w};
        const int   mtv[8] = {mtlo.x, mtlo.y, mtlo.z, mtlo.w,
                              mthi.x, mthi.y, mthi.z, mthi.w};

        // signed score per row: negative => in radius. Only floats live across
        // the rare-recheck branch (no wave-mask booleans -> no SGPR spills).
        float eff[8];
        bool anyNear = false;
        #pragma unroll
        for (int v = 0; v < 8; ++v) {
            const float delta = acc[v] + kk - r2v[v];   // dist^2 - r^2 (approx)
            eff[v] = delta;
            anyNear |= (fabsf(delta) <= MARGIN);
        }
        if (__builtin_expect(anyNear, 0)) {
            // borderline somewhere in this lane's 8 rows: exact diff-based test
            #pragma unroll
            for (int v = 0; v < 8; ++v) {
                const float4 c = sA[rbase + v];
                const float dx = kx - c.x, dy = ky - c.y, dz = kz - c.z;
                const float ex =
                    (sqrtf(dx * dx + dy * dy + dz * dz) < sRad[rbase + v])
                        ? -1.0f : 1.0f;
                eff[v] = (fabsf(eff[v]) <= MARGIN) ? ex : eff[v];
            }
        }

        #pragma unroll
        for (int v = 0; v < 8; ++v) {
            const int  m    = rbase + v;           // strictly increasing per lane
            const bool inr  = eff[v] < 0.0f;
            const int  meta = mtv[v];
            // (batch == b && class == a)  <=>  meta == (b<<2 | a): one compare
            best0 = (inr & (meta == (b4 | 0))) ? m : best0;
            best1 = (inr & (meta == (b4 | 1))) ? m : best1;
            best2 = (inr & (meta == (b4 | 2))) ? m : best2;
        }
    }

    // combine the two lane halves (rows 0-7 vs 8-15 of each tile)
    best0 = max(best0, __shfl_xor(best0, 16, 32));
    best1 = max(best1, __shfl_xor(best1, 16, 32));
    best2 = max(best2, __shfl_xor(best2, 16, 32));

    if (!hi) {
        const bool v0 = best0 >= 0, v1 = best1 >= 0, v2 = best2 >= 0;
        float4 cls4;
        cls4.x = v0 ? 1.0f : 0.0f;
        cls4.y = v1 ? 1.0f : 0.0f;
        cls4.z = v2 ? 1.0f : 0.0f;
        cls4.w = (v0 || v1 || v2) ? 0.0f : 1.0f;
        *(float4*)(out_cls + (size_t)g * (AA + 1)) = cls4;  // 16B aligned

        float r[28];
        #pragma unroll
        for (int i = 0; i < 28; ++i) r[i] = 0.0f;
        const int bests[3] = {best0, best1, best2};
        #pragma unroll
        for (int a = 0; a < AA; ++a) {
            const int idx = bests[a];
            if (idx >= 0) {
                const float* bx = &sBox[idx * 7];
                r[a * 7 + 0] = kx - bx[0];
                r[a * 7 + 1] = ky - bx[1];
                r[a * 7 + 2] = kz - bx[2];
                r[a * 7 + 3] = bx[3] / sAS[a * 3 + 0];
                r[a * 7 + 4] = bx[4] / sAS[a * 3 + 1];
                r[a * 7 + 5] = bx[5] / sAS[a * 3 + 2];
                r[a * 7 + 6] = bx[6];
            }
        }
        float4* org = (float4*)(out_reg + (size_t)g * 28);  // 112B stride, 16B aligned
        #pragma unroll
        for (int q = 0; q < 7; ++q) {
            float4 w; w.x = r[q*4+0]; w.y = r[q*4+1]; w.z = r[q*4+2]; w.w = r[q*4+3];
            org[q] = w;
        }
    }
}

extern "C" void kernel_launch(void* const* d_in, const int* in_sizes, int n_in,
                              void* d_out, int out_size, void* d_ws, size_t ws_size,
                              hipStream_t stream) {
    const float* keypoints    = (const float*)d_in[0];
    const float* boxes        = (const float*)d_in[1];
    const int*   class_ids    = (const int*)d_in[2];
    const int*   box_batch    = (const int*)d_in[3];
    const float* anchor_sizes = (const float*)d_in[4];
    const float* anchor_radii = (const float*)d_in[5];

    float* out     = (float*)d_out;
    float* out_cls = out;                                   // B*N*(A+1) floats
    float* out_reg = out + (size_t)BB * NN * (AA + 1);      // B*N*(A+1)*7 floats

    // 128 keypoints per block (8 waves x 16), exact cover of B*N = 65536
    const int blocks = (BB * NN) / 128;
    TargetAssigner_kernel<<<blocks, 256, 0, stream>>>(
        keypoints, boxes, class_ids, box_batch, anchor_sizes, anchor_radii,
        out_cls, out_reg);
}